// RepNetEmbedding_48215302865056
// MI455X (gfx1250) — compile-verified
//
#include <hip/hip_runtime.h>
#include <math.h>

typedef __attribute__((ext_vector_type(2))) float v2f;
typedef __attribute__((ext_vector_type(8))) float v8f;

#define NB 2
#define NN 512
#define ND 32
#define BIGV 1.0e10f

#define L2E 1.4426950408889634f   // log2(e)
#define LN2 0.6931471805599453f   // ln(2)

__device__ __forceinline__ void sigparams(const float* __restrict__ ra,
                                          const float* __restrict__ rb,
                                          float& a, float& b) {
  float x = ra[0];
  float elu = x > 0.f ? x : (__expf(x) - 1.f);
  a = fminf(elu + 1.f, 100.f);   // SIGMOID_A_MAX
  b = rb[0];
}

// ---------------------------------------------------------------------------
// K1: raw temporal self-similarity matrix via WMMA f32 16x16x4 rank-3 tiles.
// One BLOCK (4 waves) per 16x16 (n,m) tile; each wave handles 8 of the 32
// i-values, j fully unrolled with register-resident B fragments (no LDS in
// the hot loop). Partial v8f accumulators are combined through LDS.
// Operands are pre-scaled by log2(e) so WMMA emits z2 = z*log2e directly:
//   z2 = (L2E*-2a*x_ni)*x_mj + (L2E*(a*x_ni^2-b))*1 + 1*(L2E*a*x_mj^2)
// softplus(z) = ln2 * ( (z2+|z2|)*0.5 + log2(1 + exp2(-|z2|)) )
// ---------------------------------------------------------------------------
__global__ __launch_bounds__(128)
void tsm_kernel(const float* __restrict__ feats,
                const float* __restrict__ raw_a,
                const float* __restrict__ raw_b,
                float* __restrict__ tsm) {
  __shared__ float smem[2048 + 1024];   // 8KB operands + 4KB reduction
  const int lane = threadIdx.x & 31;
  const int wv   = threadIdx.x >> 5;

  float a, b;
  sigparams(raw_a, raw_b, a, b);

  const int tile = blockIdx.x;                          // 0..2047
  const int bb   = tile >> 10;                          // batch
  const int rem  = tile & 1023;
  const int n0   = (rem >> 5) << 4;                     // row tile * 16
  const int m0   = (rem & 31) << 4;                     // col tile * 16

  float* w_alpha = smem;                // [i=0..31][r=0..15]
  float* w_beta  = w_alpha + 512;       // [i][r]
  float* w_b0    = w_beta + 512;        // [j][lane 0..31]
  float* red     = smem + 2048;         // [wave][lane*8+r]

  const float* Xb = feats + (size_t)bb * NN * ND;
  for (int e = threadIdx.x; e < 512; e += 128) {  // n-side operands (pre-scaled)
    int r = e >> 5, i = e & 31;
    float x = Xb[(n0 + r) * ND + i];
    w_alpha[i * 16 + r] = -2.f * a * x * L2E;
    w_beta [i * 16 + r] = (a * x * x - b) * L2E;
  }
  for (int e = threadIdx.x; e < 512; e += 128) {  // m-side operands
    int r = e >> 5, j = e & 31;
    float x = Xb[(m0 + r) * ND + j];
    w_b0[j * 32 + r]      = x;                    // K=0 row (lanes 0-15)
    w_b0[j * 32 + 16 + r] = a * x * x * L2E;      // K=2 row (lanes 16-31)
  }
  __syncthreads();

  // B-side register-resident: 32 floats per lane, no LDS in the hot loop.
  float breg[32];
  #pragma unroll
  for (int j = 0; j < 32; ++j) breg[j] = w_b0[j * 32 + lane];

  const bool  lo  = lane < 16;
  const int   ls  = lane & 15;
  const float B1c = lo ? 1.f : 0.f;            // K=1 row is ones, K=3 is zero
  const v8f   cz  = {0.f, 0.f, 0.f, 0.f, 0.f, 0.f, 0.f, 0.f};
  v8f acc = cz;

  const int i0 = wv * 8;                       // this wave's i-slice
  for (int i = i0; i < i0 + 8; ++i) {
    float av = w_alpha[i * 16 + ls];
    float bv = w_beta [i * 16 + ls];
    v2f afrag;                                 // A: VGPR0={K0|K2}, VGPR1={K1|K3}
    afrag.x = lo ? av : 1.f;
    afrag.y = lo ? bv : 0.f;
    #pragma unroll
    for (int j = 0; j < 32; ++j) {
      v2f bfrag;                               // B: VGPR0={K0|K2}, VGPR1={K1|K3}
      bfrag.x = breg[j];
      bfrag.y = B1c;
      v8f z = __builtin_amdgcn_wmma_f32_16x16x4_f32(
          false, afrag, false, bfrag, (short)0, cz, false, false);
      #pragma unroll
      for (int r = 0; r < 8; ++r) {
        float zz = z[r];
        float t  = __builtin_amdgcn_exp2f(-fabsf(zz));     // v_exp_f32 -|v|
        float lg = __builtin_amdgcn_logf(1.f + t);         // v_log_f32
        // max(z,0) == (z+|z|)*0.5 ; abs folds into v_add src modifier,
        // 0.5-mul + accumulate fold into one v_fma.
        acc[r] += fmaf(zz + fabsf(zz), 0.5f, lg);
      }
    }
  }

  // Combine the 4 per-wave partials via LDS, then store the 16x16 tile.
  #pragma unroll
  for (int r = 0; r < 8; ++r) red[wv * 256 + lane * 8 + r] = acc[r];
  __syncthreads();

  const float scale = LN2 / (float)(ND * ND);  // ln2 (base-2 domain) / 1024
  float* dst = tsm + (size_t)bb * NN * NN;
  #pragma unroll
  for (int k = 0; k < 2; ++k) {
    int p = threadIdx.x + k * 128;             // flattened (lane,r) position
    float s = red[p] + red[p + 256] + red[p + 512] + red[p + 768];
    int ln  = p >> 3, r = p & 7;
    // C/D layout: VGPR r, lanes 0-15 -> (M=r, N=lane); lanes 16-31 -> (M=r+8)
    int row = n0 + ((ln >> 4) << 3) + r;
    int col = m0 + (ln & 15);
    dst[(size_t)row * NN + col] = s * scale;
  }
}

// ---------------------------------------------------------------------------
// Block reductions (256 threads)
// ---------------------------------------------------------------------------
__device__ __forceinline__ float blk_reduce_sum256(float v, float* sm) {
  int t = threadIdx.x;
  sm[t] = v; __syncthreads();
  #pragma unroll
  for (int s = 128; s > 0; s >>= 1) {
    if (t < s) sm[t] += sm[t + s];
    __syncthreads();
  }
  float r = sm[0]; __syncthreads();
  return r;
}

__device__ __forceinline__ float blk_reduce_max256(float v, float* sm) {
  int t = threadIdx.x;
  sm[t] = v; __syncthreads();
  #pragma unroll
  for (int s = 128; s > 0; s >>= 1) {
    if (t < s) sm[t] = fmaxf(sm[t], sm[t + s]);
    __syncthreads();
  }
  float r = sm[0]; __syncthreads();
  return r;
}

// ---------------------------------------------------------------------------
// K2: masked mean / unbiased std per batch (two-pass)
// ---------------------------------------------------------------------------
__global__ __launch_bounds__(256)
void stats_kernel(const float* __restrict__ tsm,
                  const int* __restrict__ masks,
                  float* __restrict__ stats) {
  __shared__ float sm[256];
  __shared__ unsigned char mk[NN];
  const int bb = blockIdx.x;
  const float* T = tsm + (size_t)bb * NN * NN;
  const int* Mk = masks + bb * NN;
  for (int i = threadIdx.x; i < NN; i += 256) mk[i] = (Mk[i] != 0) ? 1 : 0;
  __syncthreads();

  float sum = 0.f, cnt = 0.f;
  for (int idx = threadIdx.x; idx < NN * NN; idx += 256) {
    int n = idx >> 9, m = idx & (NN - 1);
    if (mk[n] & mk[m]) { sum += T[idx]; cnt += 1.f; }
  }
  float tot = blk_reduce_sum256(sum, sm);
  float c   = blk_reduce_sum256(cnt, sm);
  float mean = tot / c;

  float sq = 0.f;
  for (int idx = threadIdx.x; idx < NN * NN; idx += 256) {
    int n = idx >> 9, m = idx & (NN - 1);
    if (mk[n] & mk[m]) { float d = T[idx] - mean; sq += d * d; }
  }
  float totsq = blk_reduce_sum256(sq, sm);
  float stdv = sqrtf(totsq / (c - 1.f));

  if (threadIdx.x == 0) { stats[2 * bb] = mean; stats[2 * bb + 1] = stdv; }
}

// ---------------------------------------------------------------------------
// K3: normalize, masked fill BIG, row softmax(-x/T), zero outside mask, relu
// One block per (batch,row); 256 threads x 2 columns.
// ---------------------------------------------------------------------------
__global__ __launch_bounds__(256)
void softmax_kernel(const float* __restrict__ tsm,
                    const int* __restrict__ masks,
                    const float* __restrict__ stats,
                    float* __restrict__ out) {
  __shared__ float sm[256];
  const int blk = blockIdx.x;
  const int bb  = blk >> 9;
  const int n   = blk & (NN - 1);
  const bool rowv = masks[bb * NN + n] != 0;
  const float mean = stats[2 * bb];
  const float invstd = 1.f / stats[2 * bb + 1];
  const float invT = 1.f / 13.544f;   // TEMPERATURE
  const float* Trow = tsm + (size_t)bb * NN * NN + (size_t)n * NN;

  const int c0 = threadIdx.x, c1 = threadIdx.x + 256;
  const bool v0 = rowv && (masks[bb * NN + c0] != 0);
  const bool v1 = rowv && (masks[bb * NN + c1] != 0);
  float t0 = v0 ? (Trow[c0] - mean) * invstd : BIGV;
  float t1 = v1 ? (Trow[c1] - mean) * invstd : BIGV;
  float s0 = -t0 * invT, s1 = -t1 * invT;

  float mx = blk_reduce_max256(fmaxf(s0, s1), sm);
  float e0 = __expf(s0 - mx), e1 = __expf(s1 - mx);
  float den = blk_reduce_sum256(e0 + e1, sm);
  float inv = 1.f / den;

  float* Orow = out + (size_t)bb * NN * NN + (size_t)n * NN;
  Orow[c0] = v0 ? fmaxf(e0 * inv, 0.f) : 0.f;   // relu(where(mask, softmax, 0))
  Orow[c1] = v1 ? fmaxf(e1 * inv, 0.f) : 0.f;
}

// ---------------------------------------------------------------------------
extern "C" void kernel_launch(void* const* d_in, const int* in_sizes, int n_in,
                              void* d_out, int out_size, void* d_ws, size_t ws_size,
                              hipStream_t stream) {
  const float* feats = (const float*)d_in[0];   // [B,N,D] f32
  const int*   masks = (const int*)d_in[1];     // [B,N]   bool->int
  const float* raw_a = (const float*)d_in[2];   // [1]
  const float* raw_b = (const float*)d_in[3];   // [1]
  float* out = (float*)d_out;                   // [B,N,N] f32

  float* tsm   = (float*)d_ws;                  // B*N*N floats (2 MB)
  float* stats = tsm + (size_t)NB * NN * NN;    // 2 floats per batch

  const int tiles = NB * (NN / 16) * (NN / 16); // 2048 blocks, 4 waves each
  tsm_kernel<<<tiles, 128, 0, stream>>>(feats, raw_a, raw_b, tsm);
  stats_kernel<<<NB, 256, 0, stream>>>(tsm, masks, stats);
  softmax_kernel<<<NB * NN, 256, 0, stream>>>(tsm, masks, stats, out);
}